// DePatch_85134841741548
// MI455X (gfx1250) — compile-verified
//
#include <hip/hip_runtime.h>
#include <stdint.h>

// Geometry (baked): ORIG=(64,64,32,32,2); ST patches 8x8 step 4 (7x7); UV patches 16x16 step 8 (7x7)
// Input  x[b=1][p=2401][pu=16][pv=16][ps=8][pt=8][c=2], p = ((si*7+ti)*7+ui)*7+vi
// Output rec[1][64][64][32][32][2]

#define CHUNK_F      128          // (ps,pt,c) contiguous floats per patch-chunk
#define CHUNK_PAD_F  136          // padded LDS stride (TDM pad: +8 dwords per 128)
#define SLOT_CHUNKS  49           // si(7) x ti(7)
#define SLOT_F       (SLOT_CHUNKS * CHUNK_PAD_F)   // 6664 floats = 26,656 B

typedef unsigned int u32t;
typedef __attribute__((ext_vector_type(4))) u32t v4u_t;
typedef __attribute__((ext_vector_type(8))) int  v8i_t;
typedef __attribute__((ext_vector_type(4))) int  v4i_t;

#if defined(__HIP_DEVICE_COMPILE__) && defined(__gfx1250__) && __has_builtin(__builtin_amdgcn_tensor_load_to_lds)
#define USE_TDM 1
#else
#define USE_TDM 0
#endif

__global__ __launch_bounds__(256) void depatch_kernel(const float* __restrict__ x,
                                                      float* __restrict__ out) {
    __shared__ float smem[2 * SLOT_F];   // 53,312 bytes

    const int tid = threadIdx.x;
    const int u  = (int)blockIdx.x >> 6;  // 0..63
    const int v  = (int)blockIdx.x & 63;  // 0..63
    const int t  = tid & 31;              // 0..31
    const int s0 = tid >> 5;              // 0..7 ; thread handles s = s0 + 8k, k=0..3

    // ---- uniform (per-block) u/v covering-patch lists (<=2 each, scalar) ----
    int uis[2]; int un = 0;
    { int hi = u >> 3; if (hi <= 6) uis[un++] = hi; if (hi >= 1) uis[un++] = hi - 1; }
    int vis[2]; int vn = 0;
    { int hi = v >> 3; if (hi <= 6) vis[vn++] = hi; if (hi >= 1) vis[vn++] = hi - 1; }

    // ---- per-lane t candidates, branchless (clamped; 0/1-weighted) ----
    const int  tr      = t >> 2;
    const bool t_hi_ok = (tr <= 6);
    const bool t_lo_ok = (tr >= 1);
    const int  ti_hi   = t_hi_ok ? tr : 6;
    const int  ti_lo   = t_lo_ok ? tr - 1 : 0;
    const int  ofs_t_hi = ti_hi * CHUNK_PAD_F + (t - 4 * ti_hi) * 2;  // invariant over k/iv/iu
    const int  ofs_t_lo = ti_lo * CHUNK_PAD_F + (t - 4 * ti_lo) * 2;
    const int  tn = (int)t_hi_ok + (int)t_lo_ok;

    // ---- per-lane s candidates for the 4 s-values this thread owns (fully unrolled) ----
    int   ofs_s_hi[4], ofs_s_lo[4], sn[4];
    float w_hh[4], w_hl[4], w_lh[4], w_ll[4];   // weight[s-cand][t-cand]
#pragma unroll
    for (int k = 0; k < 4; ++k) {
        const int  s       = s0 + 8 * k;
        const int  sr      = s >> 2;
        const bool s_hi_ok = (sr <= 6);
        const bool s_lo_ok = (sr >= 1);
        const int  si_hi   = s_hi_ok ? sr : 6;
        const int  si_lo   = s_lo_ok ? sr - 1 : 0;
        ofs_s_hi[k] = si_hi * (7 * CHUNK_PAD_F) + (s - 4 * si_hi) * 16;
        ofs_s_lo[k] = si_lo * (7 * CHUNK_PAD_F) + (s - 4 * si_lo) * 16;
        sn[k] = (int)s_hi_ok + (int)s_lo_ok;
        w_hh[k] = (s_hi_ok && t_hi_ok) ? 1.f : 0.f;
        w_hl[k] = (s_hi_ok && t_lo_ok) ? 1.f : 0.f;
        w_lh[k] = (s_lo_ok && t_hi_ok) ? 1.f : 0.f;
        w_ll[k] = (s_lo_ok && t_lo_ok) ? 1.f : 0.f;
    }

    float accx[4] = {0.f, 0.f, 0.f, 0.f};
    float accy[4] = {0.f, 0.f, 0.f, 0.f};

    for (int iu = 0; iu < un; ++iu) {
        const int ui = uis[iu];
        const int pu = u - 8 * ui;

        // ---------------- stage vn slots of x[si,ti,ui,vi,pu,pv,:,:,:] into LDS ----------------
#if USE_TDM
        if (tid < 32) {  // wave 0 issues the TDM descriptors (EXEC ignored by TDM)
            const u32t lds_base = (u32t)(uintptr_t)(&smem[0]);  // generic addr low 32b = LDS byte offset
            for (int iv = 0; iv < vn; ++iv) {
                const int vi2 = vis[iv];
                const int pv  = v - 8 * vi2;
                const uint64_t off_f = (uint64_t)(ui * 7 + vi2) * 32768ull
                                     + (uint64_t)pu * 2048ull + (uint64_t)pv * 128ull;
                const uint64_t ga = (uint64_t)(uintptr_t)x + off_f * 4ull;
                const u32t lds_byte = lds_base + (u32t)(iv * SLOT_F) * 4u;

                // D# group 0: count=1 | lds_addr | global_addr[56:0] | type=2
                v4u_t g0;
                g0[0] = __builtin_amdgcn_readfirstlane(1u);
                g0[1] = __builtin_amdgcn_readfirstlane(lds_byte);
                g0[2] = __builtin_amdgcn_readfirstlane((u32t)ga);
                g0[3] = __builtin_amdgcn_readfirstlane(((u32t)(ga >> 32) & 0x1FFFFFFu) | (2u << 30));

                // D# group 1:
                //  [0] wg_mask=0 | data_size=2(4B)<<16 | pad_enable<<20 | pad_interval=6(128dw)<<22 | pad_amount=7(8dw)<<25
                //  tensor_dim0/1 = 2^30 (no OOB), tile = 128 x 7(ti) x 7(si)
                //  dim0_stride(ti) = 49*32768 = 1,605,632 elts ; dim1_stride(si) = 343*32768 = 11,239,424 elts
                v8i_t g1;
                g1[0] = (int)0x0F920000u;
                g1[1] = 0;
                g1[2] = 0x00004000;          // tensor_dim0 = 0x40000000
                g1[3] = 0x00804000;          // tensor_dim1 = 0x40000000, tile_dim0 = 128
                g1[4] = 0x00070007;          // tile_dim1 = 7, tile_dim2 = 7
                g1[5] = 0x00188000;          // dim0_stride lo (1,605,632)
                g1[6] = (int)0x80000000u;    // dim0_stride hi=0 | dim1_stride[15:0]=0x8000
                g1[7] = 0x000000AB;          // dim1_stride[47:16] (11,239,424 = 0xAB8000)

                v4i_t g2; g2[0] = 0x40000000; g2[1] = 0; g2[2] = 0; g2[3] = 0;  // tensor_dim2 big
                v4i_t g3; g3[0] = 0; g3[1] = 0; g3[2] = 0; g3[3] = 0;

                v8i_t g4;                     // extra group in the 6-arg clang-23 form (zero-filled)
                g4[0] = 0; g4[1] = 0; g4[2] = 0; g4[3] = 0;
                g4[4] = 0; g4[5] = 0; g4[6] = 0; g4[7] = 0;

                __builtin_amdgcn_tensor_load_to_lds(g0, g1, g2, g3, g4, 0);
            }
            __builtin_amdgcn_s_wait_tensorcnt(0);
        }
#else
        // Fallback: coalesced global->LDS staging with the same padded layout
        for (int iv = 0; iv < vn; ++iv) {
            const int vi2 = vis[iv];
            const int pv  = v - 8 * vi2;
            const float* src = x + ((size_t)(ui * 7 + vi2) * 32768 + (size_t)pu * 2048 + (size_t)pv * 128);
            for (int e = tid; e < SLOT_CHUNKS * (CHUNK_F / 2); e += 256) {
                const int chunk = e >> 6;          // 64 float2 per chunk
                const int r     = e & 63;
                const int si    = chunk / 7;
                const int ti2   = chunk % 7;
                const float2 val = *(const float2*)(src + ((size_t)si * 343 + (size_t)ti2 * 49) * 32768 + 2 * r);
                *(float2*)&smem[iv * SLOT_F + chunk * CHUNK_PAD_F + 2 * r] = val;
            }
        }
#endif
        __syncthreads();

        // ---------------- branchless gather: 4 conflict-free ds_load_b64 + 8 FMAs per (k,iv) ----------------
        for (int iv = 0; iv < vn; ++iv) {          // scalar-uniform loop
            const float* slot = smem + iv * SLOT_F;
#pragma unroll
            for (int k = 0; k < 4; ++k) {
                const float* ph = slot + ofs_s_hi[k];
                const float* pl = slot + ofs_s_lo[k];
                const float2 vhh = *(const float2*)(ph + ofs_t_hi);
                const float2 vhl = *(const float2*)(ph + ofs_t_lo);
                const float2 vlh = *(const float2*)(pl + ofs_t_hi);
                const float2 vll = *(const float2*)(pl + ofs_t_lo);
                float ax = accx[k], ay = accy[k];
                ax = fmaf(w_hh[k], vhh.x, ax); ay = fmaf(w_hh[k], vhh.y, ay);
                ax = fmaf(w_hl[k], vhl.x, ax); ay = fmaf(w_hl[k], vhl.y, ay);
                ax = fmaf(w_lh[k], vlh.x, ax); ay = fmaf(w_lh[k], vlh.y, ay);
                ax = fmaf(w_ll[k], vll.x, ax); ay = fmaf(w_ll[k], vll.y, ay);
                accx[k] = ax; accy[k] = ay;
            }
        }
        __syncthreads();   // LDS reused next iteration
    }

    // ---------------- normalize by overlap count and store (coalesced float2) ----------------
    float2* out2 = (float2*)out;
#pragma unroll
    for (int k = 0; k < 4; ++k) {
        const int s = s0 + 8 * k;
        const float inv = 1.0f / (float)(un * vn * sn[k] * tn);
        float2 o; o.x = accx[k] * inv; o.y = accy[k] * inv;
        out2[((size_t)((u * 64 + v) * 32 + s) * 32) + t] = o;
    }
}

extern "C" void kernel_launch(void* const* d_in, const int* in_sizes, int n_in,
                              void* d_out, int out_size, void* d_ws, size_t ws_size,
                              hipStream_t stream) {
    const float* x = (const float*)d_in[0];
    float* out = (float*)d_out;
    (void)in_sizes; (void)n_in; (void)out_size; (void)d_ws; (void)ws_size;
    depatch_kernel<<<dim3(64 * 64), dim3(256), 0, stream>>>(x, out);
}